// Reverb_43130061586544
// MI455X (gfx1250) — compile-verified
//
#include <hip/hip_runtime.h>

typedef __attribute__((ext_vector_type(2))) float v2f;
typedef __attribute__((ext_vector_type(8))) float v8f;

#define S_DIM 256
#define C_CH  32
#define O_CH  32
#define IMG   (S_DIM * S_DIM)   // 65536 pixels per (b, c) plane

// One wave32 computes a 32(out-ch) x 16(pixel) output tile:
//   D = conv_w(32x32) x h(32x16) + bias
// via 2 M-tiles x 8 K-steps of v_wmma_f32_16x16x4_f32.
__global__ __launch_bounds__(256) void reverb_fused_kernel(
    const float* __restrict__ x,      // [B, C, S, S]
    const float* __restrict__ wgt,    // [1, C, S, S]
    const float* __restrict__ conv_w, // [O, C] row-major
    const float* __restrict__ conv_b, // [O]
    float* __restrict__ out)          // [B, O, S, S]
{
    const int lane = threadIdx.x & 31;
    const int wave = threadIdx.x >> 5;
    const int tile = blockIdx.x * 8 + wave;   // 16-pixel tile id, 65536 total
    const int n    = lane & 15;               // pixel column within tile
    const int hi   = lane >> 4;               // half-wave selector

    const int b     = tile >> 12;             // 4096 tiles per image
    const int pbase = (tile & 4095) << 4;     // pixel base within image
    const int p     = pbase + n;              // this lane's pixel
    const int row   = p >> 8;
    const int col   = p & 255;

    // Separable overlap count: c1(s) = 5 - max(0,2-s) - max(0,s-253)
    const int c1r = 5 - max(0, 2 - row) - max(0, row - 253);
    const int c1c = 5 - max(0, 2 - col) - max(0, col - 253);
    const float cnt = (float)(c1r * c1c);

    // ---- A fragments (conv_w), 16x4 f32 layout ----
    // lane L: row M = L&15 ; VGPR0/VGPR1 hold K = 4k + 2*hi, 4k + 2*hi + 1
    v2f a[2][8];
    {
        const int mrow = lane & 15;
#pragma unroll
        for (int m = 0; m < 2; ++m) {
            const int arow = m * 16 + mrow;
#pragma unroll
            for (int k = 0; k < 8; ++k) {
                const int cb = 4 * k + 2 * hi;
                a[m][k] = *(const v2f*)(conv_w + arow * C_CH + cb);
            }
        }
    }

    // ---- B fragments: h[c, n] = x*weight*count for this lane's 2 channels/K-step ----
    const float* xp = x + (size_t)b * C_CH * IMG + p;
    const float* wp = wgt + p;
    float hv[16];
#pragma unroll
    for (int k = 0; k < 8; ++k) {
        const int c0 = 4 * k + 2 * hi;
        hv[2 * k]     = xp[(size_t)c0 * IMG]       * wp[(size_t)c0 * IMG]       * cnt;
        hv[2 * k + 1] = xp[(size_t)(c0 + 1) * IMG] * wp[(size_t)(c0 + 1) * IMG] * cnt;
    }

    // ---- Accumulators initialized with bias (free epilogue) ----
    // C/D layout: VGPR i -> row M = i + 8*hi (per 16-row tile), col N = lane&15
    v8f acc0, acc1;
#pragma unroll
    for (int i = 0; i < 8; ++i) {
        acc0[i] = conv_b[i + 8 * hi];
        acc1[i] = conv_b[16 + i + 8 * hi];
    }

    // ---- 8 K-steps x 2 M-tiles of f32 WMMA ----
#pragma unroll
    for (int k = 0; k < 8; ++k) {
        v2f bk;
        bk[0] = hv[2 * k];
        bk[1] = hv[2 * k + 1];
        acc0 = __builtin_amdgcn_wmma_f32_16x16x4_f32(
            false, a[0][k], false, bk, (short)0, acc0, false, false);
        acc1 = __builtin_amdgcn_wmma_f32_16x16x4_f32(
            false, a[1][k], false, bk, (short)0, acc1, false, false);
    }

    // ---- Store: per VGPR i, half-wave writes 64 contiguous bytes of channel o ----
    float* op = out + (size_t)b * O_CH * IMG + pbase + n;
#pragma unroll
    for (int i = 0; i < 8; ++i) {
        op[(size_t)(i + 8 * hi) * IMG]      = acc0[i];
        op[(size_t)(16 + i + 8 * hi) * IMG] = acc1[i];
    }
}

extern "C" void kernel_launch(void* const* d_in, const int* in_sizes, int n_in,
                              void* d_out, int out_size, void* d_ws, size_t ws_size,
                              hipStream_t stream) {
    const float* x      = (const float*)d_in[0]; // [16,32,256,256]
    const float* weight = (const float*)d_in[1]; // [1,32,256,256]
    const float* conv_w = (const float*)d_in[2]; // [32,32]
    const float* conv_b = (const float*)d_in[3]; // [32]
    float* out = (float*)d_out;                  // [16,32,256,256]

    // 65536 16-pixel tiles, 8 waves (tiles) per 256-thread block
    const int blocks = 65536 / 8;
    reverb_fused_kernel<<<blocks, 256, 0, stream>>>(x, weight, conv_w, conv_b, out);
}